// PoseMapGenerator_41901700939945
// MI455X (gfx1250) — compile-verified
//
#include <hip/hip_runtime.h>
#include <hip/hip_bf16.h>

// Problem constants (from the reference)
#define NB    16          // B
#define NP    17          // P (parts)
#define NH    192         // H
#define NW    192         // W
#define MAXP  8
#define NPER  7           // fori_loop(0, MAXP-1): persons 0..6 only
#define TS    9           // template side (6*SIGMA+3)
#define RR    4           // radius offset (3*SIGMA+1)
#define HW    (NH * NW)          // 36864 pixels per plane
#define NGRP  (HW / 4)           // 9216 float4 groups per plane
#define GX    (NGRP / 256)       // 36 blocks along x per plane
#define GROUPS_PER_ROW (NW / 4)  // 48

__global__ __launch_bounds__(256)
void posemap_gather_kernel(const float* __restrict__ kt,     // (9,9) gaussian*disk template
                           const float* __restrict__ emb,    // (MAXP, B) embeddings
                           const int*   __restrict__ kpf,    // (N,2) keypoints (x,y)
                           float* __restrict__ out)          // (B, 2P, H, W)
{
    __shared__ float s_tmpl[TS * TS];    // 81 dwords

    const int tid = threadIdx.x;
    const int bp  = blockIdx.y;          // 0..271
    const int b   = bp / NP;             // batch
    const int pid = bp - b * NP;         // part

    // ---- CDNA5 async copy (ASYNCcnt): template global -> LDS, one b32 per lane ----
    if (tid < TS * TS) {
        const float* src     = kt + tid;
        unsigned     ldsaddr = (unsigned)(uintptr_t)(&s_tmpl[tid]);
        asm volatile("global_load_async_to_lds_b32 %0, %1, off"
                     :: "v"(ldsaddr), "v"(src)
                     : "memory");
    }

    // ---- per-block-uniform person data -> registers (scalar loads) ----
    int   kx[NPER], ky[NPER];
    float ke[NPER];
#pragma unroll
    for (int i = 0; i < NPER; ++i) {
        const int n = (b * MAXP + i) * NP + pid;   // keypoint row for (b, person i, pid)
        kx[i] = kpf[2 * n];
        ky[i] = kpf[2 * n + 1];
        ke[i] = emb[i * NB + b];
    }

    asm volatile("s_wait_asynccnt 0x0" ::: "memory");
    __syncthreads();

    // ---- gather: 4 consecutive pixels per lane ----
    const int g  = blockIdx.x * 256 + tid;          // float4 group id within plane, 0..9215
    const int r  = g / GROUPS_PER_ROW;              // row 0..191
    const int c0 = (g - r * GROUPS_PER_ROW) * 4;    // column of first pixel

    float kv0 = 0.f, kv1 = 0.f, kv2 = 0.f, kv3 = 0.f;  // running max of template
    float ev0 = 0.f, ev1 = 0.f, ev2 = 0.f, ev3 = 0.f;  // embedding of first argmax

#pragma unroll
    for (int i = 0; i < NPER; ++i) {
        const int dy = r - ky[i];
        if ((unsigned)(dy + RR) <= (unsigned)(2 * RR)) {   // |dy| <= 4 (rare: ~4.7%/person)
            const float e      = ke[i];
            const int   rowoff = (dy + RR) * TS;
            const int   d0     = c0 - kx[i] + RR;          // dx+4 for pixel 0
            if ((unsigned)(d0 + 0) <= 8u) {
                float t = s_tmpl[rowoff + d0 + 0];
                if (t > kv0) { kv0 = t; ev0 = e; }
            }
            if ((unsigned)(d0 + 1) <= 8u) {
                float t = s_tmpl[rowoff + d0 + 1];
                if (t > kv1) { kv1 = t; ev1 = e; }
            }
            if ((unsigned)(d0 + 2) <= 8u) {
                float t = s_tmpl[rowoff + d0 + 2];
                if (t > kv2) { kv2 = t; ev2 = e; }
            }
            if ((unsigned)(d0 + 3) <= 8u) {
                float t = s_tmpl[rowoff + d0 + 3];
                if (t > kv3) { kv3 = t; ev3 = e; }
            }
        }
    }

    // ---- b128 stores: keypoint channel pid, embedding channel pid+17 ----
    const int    p     = g * 4;
    const size_t basek = (size_t)(b * (2 * NP) + pid) * (size_t)HW + (size_t)p;
    const size_t basee = basek + (size_t)NP * (size_t)HW;

    *reinterpret_cast<float4*>(out + basek) = make_float4(kv0, kv1, kv2, kv3);
    *reinterpret_cast<float4*>(out + basee) = make_float4(ev0, ev1, ev2, ev3);
}

extern "C" void kernel_launch(void* const* d_in, const int* in_sizes, int n_in,
                              void* d_out, int out_size, void* d_ws, size_t ws_size,
                              hipStream_t stream)
{
    (void)in_sizes; (void)n_in; (void)out_size; (void)d_ws; (void)ws_size;

    const float* kt  = (const float*)d_in[0];  // keypoints_temp (9,9)
    // d_in[1] embedding_temp: implied by kt>0 (disk), not needed
    const float* emb = (const float*)d_in[2];  // embeddings (8,16)
    // d_in[3..5] batch/person/part ids are derivable from index layout
    const int*   kpf = (const int*)d_in[6];    // keypoints_flat (2176,2)

    dim3 grid(GX, NB * NP, 1);                 // (36, 272)
    posemap_gather_kernel<<<grid, 256, 0, stream>>>(kt, emb, kpf, (float*)d_out);
}